// OBELISKNet_66245575573954
// MI455X (gfx1250) — compile-verified
//
#include <hip/hip_runtime.h>

typedef float v2f __attribute__((ext_vector_type(2)));
typedef float v8f __attribute__((ext_vector_type(8)));

#define DSZ   96        // D == H == W == 96
#define CCH   16        // channels
#define FEATS 512
#define KK    1024
#define NB    2         // batch

// ---------------------------------------------------------------------------
// Kernel 1: transpose [B,C,D,H,W] -> [B,D,H,W,C] so the 16 channels at one
// voxel are a single contiguous 64B line (one cacheline-friendly gather per
// trilinear corner instead of 16 reads 3.4MB apart).
// ---------------------------------------------------------------------------
__global__ void __launch_bounds__(256)
obelisk_transpose(const float* __restrict__ src, float* __restrict__ dst, int total) {
    int idx = blockIdx.x * blockDim.x + threadIdx.x;   // idx = ((b*D+z)*H+y)*W+x
    if (idx >= total) return;
    int b = idx / (DSZ * DSZ * DSZ);
    const size_t dhw = (size_t)DSZ * DSZ * DSZ;
    const float* s = src + ((size_t)b * CCH) * dhw + (size_t)(idx - b * (int)dhw);
    float vals[CCH];
#pragma unroll
    for (int c = 0; c < CCH; ++c) vals[c] = s[(size_t)c * dhw];
    float4* d = (float4*)(dst + (size_t)idx * CCH);
#pragma unroll
    for (int q = 0; q < 4; ++q)
        d[q] = make_float4(vals[4*q], vals[4*q+1], vals[4*q+2], vals[4*q+3]);
}

// ---------------------------------------------------------------------------
// Kernel 2: one workgroup per sample n = b*FEATS + f.
//   Phase 1: off[g] = A_n(3x3) @ weight[3g:3g+3] (3x1024) via
//            V_WMMA_F32_16X16X4_F32 tiles (16 k-columns per WMMA), -> LDS.
//   Phase 2: each thread processes 4 k's: two trilinear samples (zero pad,
//            align_corners=False) over 16 channels from the channel-last
//            volume, accumulating +grid1 / -grid2.
// ---------------------------------------------------------------------------
__global__ void __launch_bounds__(256)
obelisk_main(const float* __restrict__ volT,
             const float* __restrict__ xyz,
             const float* __restrict__ Amat,
             const float* __restrict__ weight,
             float* __restrict__ out) {
    __shared__ float lds_off[6][KK];                  // rows: g*3+i, 24 KB

    const int n    = blockIdx.x;                      // 0 .. NB*FEATS-1
    const int b    = n / FEATS;
    const int f    = n % FEATS;
    const int tid  = threadIdx.x;
    const int lane = tid & 31;
    const int wave = tid >> 5;
    const int m    = lane & 15;                       // row within half-wave
    const int hi   = lane >> 4;                       // which K pair

    // ---- Phase 1: offsets via WMMA --------------------------------------
    // A operand (32-bit 16x4): VGPR0 = K0|K2, VGPR1 = K1|K3 split at lane 16.
    // Only rows 0..2, cols 0..2 are live (A_n); rest zero-padded.
    const float* An = Amat + (size_t)n * 9;
    v2f aop; aop.x = 0.f; aop.y = 0.f;
    if (m < 3) {
        aop.x = An[m * 3 + (hi ? 2 : 0)];
        aop.y = hi ? 0.f : An[m * 3 + 1];
    }

    for (int t = wave; t < KK / 16; t += 8) {         // uniform trip count/wave
        const int k0 = t * 16;
#pragma unroll
        for (int g = 0; g < 2; ++g) {
            const float* w0 = weight + (size_t)(g * 3) * KK;   // rows 3g..3g+2
            const int col = k0 + m;
            // B operand (32-bit 4x16), mirrored layout; row 3 zero-padded.
            v2f bop;
            bop.x = hi ? w0[2 * KK + col] : w0[col];
            bop.y = hi ? 0.f : w0[KK + col];
            v8f acc = {};
            acc = __builtin_amdgcn_wmma_f32_16x16x4_f32(
                /*neg_a=*/false, aop, /*neg_b=*/false, bop,
                /*c_mod=*/(short)0, acc, /*reuse_a=*/false, /*reuse_b=*/false);
            // D rows 0..2 live on lanes 0..15 in VGPRs 0..2
            if (lane < 16) {
                lds_off[g * 3 + 0][k0 + lane] = acc[0];
                lds_off[g * 3 + 1][k0 + lane] = acc[1];
                lds_off[g * 3 + 2][k0 + lane] = acc[2];
            }
        }
    }
    __syncthreads();

    // ---- Phase 2: trilinear gathers -------------------------------------
    const float sx = xyz[(size_t)n * 3 + 0];
    const float sy = xyz[(size_t)n * 3 + 1];
    const float sz = xyz[(size_t)n * 3 + 2];
    const float* vb = volT + (size_t)b * (DSZ * DSZ * DSZ) * CCH;

    for (int k = tid; k < KK; k += 256) {
        float acc[CCH];
#pragma unroll
        for (int c = 0; c < CCH; ++c) acc[c] = 0.f;

#pragma unroll
        for (int g = 0; g < 2; ++g) {
            const float sign = g ? -1.f : 1.f;
            // torch stacking: grid[...,0]=zf->W, [...,1]=yf->H, [...,2]=xf->D
            const float cx = sx + lds_off[g * 3 + 2][k];   // W axis
            const float cy = sy + lds_off[g * 3 + 1][k];   // H axis
            const float cz = sz + lds_off[g * 3 + 0][k];   // D axis
            const float px = (cx + 1.f) * (0.5f * DSZ) - 0.5f;
            const float py = (cy + 1.f) * (0.5f * DSZ) - 0.5f;
            const float pz = (cz + 1.f) * (0.5f * DSZ) - 0.5f;
            const float fx = floorf(px), fy = floorf(py), fz = floorf(pz);
            const int ix = (int)fx, iy = (int)fy, iz = (int)fz;
            const float tx = px - fx, ty = py - fy, tz = pz - fz;
            const float wx[2] = {1.f - tx, tx};
            const float wy[2] = {1.f - ty, ty};
            const float wz[2] = {1.f - tz, tz};

#pragma unroll
            for (int dz = 0; dz < 2; ++dz)
#pragma unroll
                for (int dy = 0; dy < 2; ++dy)
#pragma unroll
                    for (int dx = 0; dx < 2; ++dx) {
                        const int xi = ix + dx, yi = iy + dy, zi = iz + dz;
                        if ((unsigned)xi < (unsigned)DSZ &&
                            (unsigned)yi < (unsigned)DSZ &&
                            (unsigned)zi < (unsigned)DSZ) {
                            const float w = sign * wx[dx] * wy[dy] * wz[dz];
                            const float4* p = (const float4*)
                                (vb + ((((size_t)zi * DSZ + yi) * DSZ + xi) * CCH));
#pragma unroll
                            for (int q = 0; q < 4; ++q) {
                                const float4 v = p[q];   // 64B contiguous line
                                acc[4*q+0] += w * v.x;
                                acc[4*q+1] += w * v.y;
                                acc[4*q+2] += w * v.z;
                                acc[4*q+3] += w * v.w;
                            }
                        }
                    }
        }

        // out[((b*C + c)*FEATS + f)*KK + k]; consecutive lanes -> consecutive k
        float* ob = out + ((size_t)b * CCH) * FEATS * KK + (size_t)f * KK + k;
#pragma unroll
        for (int c = 0; c < CCH; ++c)
            ob[(size_t)c * FEATS * KK] = acc[c];
    }
}

// ---------------------------------------------------------------------------
extern "C" void kernel_launch(void* const* d_in, const int* in_sizes, int n_in,
                              void* d_out, int out_size, void* d_ws, size_t ws_size,
                              hipStream_t stream) {
    const float* vol    = (const float*)d_in[0];   // [2,16,96,96,96] f32
    const float* xyz    = (const float*)d_in[1];   // [2,512,3]       f32
    const float* Amat   = (const float*)d_in[2];   // [1024,3,3]      f32
    const float* weight = (const float*)d_in[3];   // [6,1024]        f32
    float* out  = (float*)d_out;                   // [2, 16*512, 1024] f32
    float* volT = (float*)d_ws;                    // channel-last volume, 113 MB

    const int total = NB * DSZ * DSZ * DSZ;
    obelisk_transpose<<<(total + 255) / 256, 256, 0, stream>>>(vol, volT, total);
    obelisk_main<<<NB * FEATS, 256, 0, stream>>>(volT, xyz, Amat, weight, out);
}